// IRevRNN_24077586661529
// MI455X (gfx1250) — compile-verified
//
#include <hip/hip_runtime.h>
#include <math.h>

// irevrnn: elementwise reversible scan. (S,B,H)=(2048,32,1024), REV_LEN=16.
// Memory-bound: 512MB traffic -> ~22us floor @ 23.3TB/s. Parallelism is
// hard-capped at B*H = 32768 independent recurrences = 1024 wave32s
// (~1 wave/SIMD), so per-wave memory-level parallelism is the only latency
// lever: 4-stage rotating register pipeline keeps ~48-64 dword loads in
// flight per wave (the full 6-bit LOADcnt window) + global_prefetch_b8
// warming L2 4 tiles ahead of the load frontier. tanh -> v_tanh_f32
// WMMA inapplicable: no contraction dim.

#define SEQ_LEN 2048
#define BATCH   32
#define HIDDEN  1024
#define REV_LEN 16
#define NE      (BATCH * HIDDEN)      // 32768 element-recurrences
#define TILE    REV_LEN               // 16-step tiles align with weight cycle
#define NTILES  (SEQ_LEN / TILE)      // 128
#define BLOCK   128                   // 4 wave32 per WG -> 256 WGs

__device__ __forceinline__ float fast_tanh(float x) {
#if __has_builtin(__builtin_amdgcn_tanhf)
    return __builtin_amdgcn_tanhf(x);            // gfx1250 v_tanh_f32
#elif __has_builtin(__builtin_amdgcn_tanh_f32)
    return __builtin_amdgcn_tanh_f32(x);
#else
    float e = __expf(x + x);
    return 1.0f - __fdividef(2.0f, e + 1.0f);
#endif
}

// Load one 16-step tile of z into a register buffer, advance load frontier.
#define LOADT(buf)                                                         \
    do {                                                                   \
        _Pragma("unroll")                                                  \
        for (int k = 0; k < TILE; ++k)                                     \
            buf[k] = __builtin_nontemporal_load(zl + (size_t)k * NE);      \
        zl += (size_t)TILE * NE;                                           \
    } while (0)

// Run 16 recurrence steps from a register buffer, stream h out, advance.
#define COMPT(buf)                                                         \
    do {                                                                   \
        _Pragma("unroll")                                                  \
        for (int k = 0; k < TILE; ++k) {                                   \
            float a = fmaxf(iw * buf[k], 0.0f);                            \
            c += fast_tanh(fmaf(hw[k], h, a));                             \
            h += fast_tanh(cw[k] * c);                                     \
            __builtin_nontemporal_store(h, ot + (size_t)k * NE);           \
        }                                                                  \
        ot += (size_t)TILE * NE;                                           \
    } while (0)

__global__ __launch_bounds__(BLOCK, 1) void irevrnn_scan(
    const float* __restrict__ z,     // (S,B,H)
    const float* __restrict__ h0,    // (B,H)
    const float* __restrict__ c0,    // (B,H)
    const float* __restrict__ iw_,   // (1,H)
    const float* __restrict__ hw_,   // (16,H)
    const float* __restrict__ cw_,   // (16,H)
    float* __restrict__ out)         // (S,B,H)
{
    const int gid = blockIdx.x * BLOCK + threadIdx.x;   // 0..NE-1
    const int hid = gid & (HIDDEN - 1);

    // Per-element weights in registers; compile-time indexed only.
    float hw[REV_LEN], cw[REV_LEN];
#pragma unroll
    for (int k = 0; k < REV_LEN; ++k) {
        hw[k] = hw_[k * HIDDEN + hid];
        cw[k] = cw_[k * HIDDEN + hid];
    }
    const float iw = iw_[hid];
    float h = h0[gid];
    float c = c0[gid];

    const float* zl = z + gid;     // load frontier (advances per tile)
    float*       ot = out + gid;   // store frontier

    // 4-stage rotating pipeline: while computing tile t, tiles t+1..t+3
    // are in flight (up to 64 outstanding dword loads per wave).
    float za[TILE], zb[TILE], zc[TILE], zd[TILE];
    LOADT(za);
    LOADT(zb);
    LOADT(zc);

#pragma unroll 1
    for (int st = 0; st < NTILES / 4 - 1; ++st) {       // 31 iterations
        // L2 warm 4 tiles past the load frontier (max tile 127: in bounds).
        __builtin_prefetch(zl + (size_t)4 * TILE * NE, 0, 0);
        LOADT(zd); COMPT(za);
        LOADT(za); COMPT(zb);
        LOADT(zb); COMPT(zc);
        LOADT(zc); COMPT(zd);
    }
    // Epilogue: buffers hold tiles 124..126, load 127, drain all four.
    LOADT(zd);
    COMPT(za);
    COMPT(zb);
    COMPT(zc);
    COMPT(zd);
}

extern "C" void kernel_launch(void* const* d_in, const int* in_sizes, int n_in,
                              void* d_out, int out_size, void* d_ws, size_t ws_size,
                              hipStream_t stream) {
    (void)in_sizes; (void)n_in; (void)out_size; (void)d_ws; (void)ws_size;
    const float* z  = (const float*)d_in[0];
    const float* h0 = (const float*)d_in[1];
    const float* c0 = (const float*)d_in[2];
    const float* iw = (const float*)d_in[3];
    const float* hw = (const float*)d_in[4];
    const float* cw = (const float*)d_in[5];
    float* out = (float*)d_out;

    dim3 grid(NE / BLOCK);   // 256 workgroups x 128 threads = 32768 threads
    dim3 block(BLOCK);
    irevrnn_scan<<<grid, block, 0, stream>>>(z, h0, c0, iw, hw, cw, out);
}